// SoftDTWTorch_45329084842366
// MI455X (gfx1250) — compile-verified
//
#include <hip/hip_runtime.h>

#define T_N 4096
#define LOG2E 1.4426950408889634f
#define LN2   0.6931471805599453f
#define BIGS  1.4426950e10f          /* 1e10 scaled into log2 domain */
#define NEG2LOG2E (-2.8853900817779268f)

typedef __attribute__((ext_vector_type(2))) float v2f;
typedef __attribute__((ext_vector_type(8))) float v8f;

// ---------------------------------------------------------------------------
// Kernel 1: row squared-norms of A and B, pre-scaled by log2(e).
// ---------------------------------------------------------------------------
__global__ void sdtw_norms_kernel(const float* __restrict__ A,
                                  const float* __restrict__ B,
                                  float* __restrict__ a2,
                                  float* __restrict__ b2) {
    int idx = blockIdx.x * blockDim.x + threadIdx.x;
    const float* src;
    float* dst;
    int r;
    if (idx < T_N) { src = A; dst = a2; r = idx; }
    else           { src = B; dst = b2; r = idx - T_N; }
    const float4* p = (const float4*)(src + (size_t)r * 16);
    float s = 0.0f;
#pragma unroll
    for (int q = 0; q < 4; ++q) {
        float4 v = p[q];
        s += v.x * v.x + v.y * v.y + v.z * v.z + v.w * v.w;
    }
    dst[r] = s * LOG2E;
}

// ---------------------------------------------------------------------------
// Kernel 2: Dm[i,j] = log2e * (a2[i] + b2[j] - 2*(A B^T)[i,j])
// via V_WMMA_F32_16X16X4_F32 (K=16 -> 4 chained k=4 WMMAs).
// One wave per 16x16 tile, 8 waves per block (128 rows x 16 cols).
// ---------------------------------------------------------------------------
__global__ __launch_bounds__(256)
void sdtw_dist_kernel(const float* __restrict__ A,
                      const float* __restrict__ B,
                      const float* __restrict__ a2,
                      const float* __restrict__ b2,
                      float* __restrict__ Dm) {
    const int wave = threadIdx.x >> 5;
    const int lane = threadIdx.x & 31;
    const int half = lane >> 4;
    const int r    = lane & 15;

    const int m0 = blockIdx.x * 128 + wave * 16;
    const int n0 = blockIdx.y * 16;

    // Fragment loads: X[row0 + r, 4q + 2*half + v]  (float2, 8B aligned)
    const float* arow = A + (size_t)(m0 + r) * 16 + 2 * half;
    const float* brow = B + (size_t)(n0 + r) * 16 + 2 * half;

    v8f acc = {0.f, 0.f, 0.f, 0.f, 0.f, 0.f, 0.f, 0.f};
#pragma unroll
    for (int q = 0; q < 4; ++q) {
        v2f a, b;
        a.x = arow[4 * q + 0];
        a.y = arow[4 * q + 1];
        b.x = brow[4 * q + 0];
        b.y = brow[4 * q + 1];
        acc = __builtin_amdgcn_wmma_f32_16x16x4_f32(
            false, a, false, b, (short)0, acc, false, false);
    }

    const float b2v = b2[n0 + r];     // already log2e-scaled
#pragma unroll
    for (int g = 0; g < 8; ++g) {
        const int m = m0 + g + 8 * half;
        const float val = fmaf(NEG2LOG2E, acc[g], a2[m] + b2v);
        Dm[(size_t)m * T_N + (n0 + r)] = val;
    }
}

// ---------------------------------------------------------------------------
// DP step: compute diagonal k (dst) from diagonals k-1 (p1) and k-2 (p2).
// All values are in the log2 domain (scaled by log2e).
// ---------------------------------------------------------------------------
__device__ __forceinline__ void dp_step(
    int k, int t, int wbase,
    const float* __restrict__ dk,       // = dbase + (k - 2)
    float (&p1)[4], float (&p2)[4], float (&dst)[4],
    float (*__restrict__ s1)[1024], float (*__restrict__ s2)[1024])
{
    const int kb = k & 1;
    float pm1, pm2;                     // boundary values at row index 4t
    if (t == 0) {
        pm1 = BIGS;                     // R[0, k-1], k-1 >= 1 -> BIG
        pm2 = (k == 2) ? 0.0f : BIGS;   // R[0, k-2]: only R[0,0] = 0
    } else {
        pm1 = s1[kb ^ 1][t - 1];
        pm2 = s2[kb ^ 1][t - 1];
    }

    const int w = k - wbase;            // (j-1) for cell c is (w - c)

    // Anti-diagonal D loads: provably in [0, T*T) for all k, so no clamping.
    float dd[4];
    dd[0] = dk[0];
    dd[1] = dk[1 * (T_N - 1)];
    dd[2] = dk[2 * (T_N - 1)];
    dd[3] = dk[3 * (T_N - 1)];

#pragma unroll
    for (int c = 0; c < 4; ++c) {
        const float up   = (c == 0) ? pm1 : p1[c - 1];   // R[i-1, j]
        const float diag = (c == 0) ? pm2 : p2[c - 1];   // R[i-1, j-1]
        const float left = p1[c];                        // R[i, j-1]
        const float m = fminf(up, fminf(diag, left));
        const float s = __builtin_amdgcn_exp2f(m - up)
                      + __builtin_amdgcn_exp2f(m - diag)
                      + __builtin_amdgcn_exp2f(m - left);
        const float rr = dd[c] + m - __builtin_amdgcn_logf(s);
        const bool valid = (unsigned)(w - c) < (unsigned)T_N;
        dst[c] = valid ? rr : BIGS;
    }

    s1[kb][t] = dst[3];                 // diag k   boundary
    s2[kb][t] = p1[3];                  // diag k-1 boundary
    __syncthreads();
}

// ---------------------------------------------------------------------------
// Kernel 3: soft-DTW anti-diagonal DP.  One workgroup (32 wave32s) on a
// single WGP; 4 rows per thread held in registers; one barrier + 4 LDS ops
// per thread per diagonal; k-loop unrolled by 3 so the r2<-r1<-rn rotation
// is pure register renaming.
// ---------------------------------------------------------------------------
__global__ __launch_bounds__(1024, 1)
void sdtw_dp_kernel(const float* __restrict__ Dm, float* __restrict__ out) {
    __shared__ float s1[2][1024];
    __shared__ float s2[2][1024];

    const int t = threadIdx.x;
    const int wbase = 4 * t + 2;
    const float* dbase = Dm + (size_t)(4 * t) * (T_N - 1);

    float r1v[4], r2v[4], rn[4];
#pragma unroll
    for (int c = 0; c < 4; ++c) { r1v[c] = BIGS; r2v[c] = BIGS; }
    s1[1][t] = BIGS;                    // k=2 reads buffer 1
    s2[1][t] = BIGS;
    __syncthreads();

    int k = 2;
    for (; k + 2 <= 2 * T_N; k += 3) {  // 2730 triples: k = 2 .. 8191
        dp_step(k,     t, wbase, dbase + (k - 2), r1v, r2v, rn,  s1, s2);
        dp_step(k + 1, t, wbase, dbase + (k - 1), rn,  r1v, r2v, s1, s2);
        dp_step(k + 2, t, wbase, dbase + (k    ), r2v, rn,  r1v, s1, s2);
    }
    // tail: k == 2*T_N == 8192
    dp_step(k, t, wbase, dbase + (k - 2), r1v, r2v, rn, s1, s2);

    if (t == 1023) out[0] = rn[3] * LN2;   // R[T, T] back to natural domain
}

// ---------------------------------------------------------------------------
extern "C" void kernel_launch(void* const* d_in, const int* in_sizes, int n_in,
                              void* d_out, int out_size, void* d_ws, size_t ws_size,
                              hipStream_t stream) {
    (void)in_sizes; (void)n_in; (void)out_size; (void)ws_size;
    const float* A = (const float*)d_in[0];
    const float* B = (const float*)d_in[1];
    float* out = (float*)d_out;

    float* ws = (float*)d_ws;
    float* Dm = ws;                                   // T*T floats (64 MB)
    float* a2 = ws + (size_t)T_N * T_N;               // T floats
    float* b2 = a2 + T_N;                             // T floats

    sdtw_norms_kernel<<<(2 * T_N) / 256, 256, 0, stream>>>(A, B, a2, b2);

    dim3 grid(T_N / 128, T_N / 16);                   // (32, 256)
    sdtw_dist_kernel<<<grid, 256, 0, stream>>>(A, B, a2, b2, Dm);

    sdtw_dp_kernel<<<1, 1024, 0, stream>>>(Dm, out);
}